// DecodeAttnModel_7670811591162
// MI455X (gfx1250) — compile-verified
//
#include <hip/hip_runtime.h>
#include <hip/hip_bf16.h>

#define B_      4
#define H_      128
#define HID_    7168
#define QLORA_  1536
#define KVLORA_ 512
#define DNOPE_  128
#define DROPE_  64
#define DV_     128
#define DQK_    192
#define PAST_   1024
#define TTOT_   1025
#define CW_     576     // KVLORA_ + DROPE_
#define EPS_    1e-6f

typedef __attribute__((ext_vector_type(2))) float v2f;
typedef __attribute__((ext_vector_type(8))) float v8f;
typedef int b128_t __attribute__((vector_size(16)));   // 16-byte chunk for async LDS copy

#if defined(__has_builtin)
#  if __has_builtin(__builtin_amdgcn_global_load_async_to_lds_b128) && \
      __has_builtin(__builtin_amdgcn_s_wait_asynccnt)
#    define ASYNC_LDS_OK 1
#  endif
#endif

__device__ __forceinline__ float wave_sum(float v) {
#pragma unroll
    for (int off = 16; off; off >>= 1) v += __shfl_xor(v, off, 32);
    return v;
}
__device__ __forceinline__ float wave_max(float v) {
#pragma unroll
    for (int off = 16; off; off >>= 1) v = fmaxf(v, __shfl_xor(v, off, 32));
    return v;
}

// ---------------------------------------------------------------------------
// Stage 1: q_a_raw = x @ w_qa.T ; ckv_raw = x @ w_kva.T   (one wave per output)
// ---------------------------------------------------------------------------
__global__ void k_gemv_qa_kva(const float* __restrict__ x,
                              const float* __restrict__ w_qa,
                              const float* __restrict__ w_kva,
                              float* __restrict__ qa_raw,
                              float* __restrict__ ckv_raw) {
    const int wid  = (blockIdx.x * blockDim.x + threadIdx.x) >> 5;
    const int lane = threadIdx.x & 31;
    const int per_b = QLORA_ + CW_;              // 2112
    const int b = wid / per_b;
    const int o = wid % per_b;
    if (b >= B_) return;
    const float* xb = x + (size_t)b * HID_;
    const float* row = (o < QLORA_) ? (w_qa + (size_t)o * HID_)
                                    : (w_kva + (size_t)(o - QLORA_) * HID_);
    float acc = 0.f;
    for (int i = lane * 4; i < HID_; i += 128) {
        float4 xv = *(const float4*)(xb + i);
        float4 wv = *(const float4*)(row + i);
        acc += xv.x * wv.x + xv.y * wv.y + xv.z * wv.z + xv.w * wv.w;
    }
    acc = wave_sum(acc);
    if (lane == 0) {
        if (o < QLORA_) qa_raw[b * QLORA_ + o] = acc;
        else            ckv_raw[b * CW_ + (o - QLORA_)] = acc;
    }
}

// ---------------------------------------------------------------------------
// Stage 2: RMS-norm q_a and c_kv, RoPE k_pe, write new latent row into cache
// ---------------------------------------------------------------------------
__global__ void k_norm_rope_cache(const float* __restrict__ qa_raw,
                                  const float* __restrict__ ckv_raw,
                                  const float* __restrict__ gamma_q,
                                  const float* __restrict__ gamma_kv,
                                  const long long* __restrict__ pos_ids,
                                  const float* __restrict__ cosT,
                                  const float* __restrict__ sinT,
                                  float* __restrict__ q_norm,
                                  float* __restrict__ cache_out) {
    const int b = blockIdx.x;
    const int tid = threadIdx.x;             // 256 threads
    __shared__ float red[256];

    // rms(q_a)
    float ss = 0.f;
    for (int i = tid; i < QLORA_; i += 256) {
        float v = qa_raw[b * QLORA_ + i];
        ss += v * v;
    }
    red[tid] = ss;
    __syncthreads();
    for (int s = 128; s; s >>= 1) { if (tid < s) red[tid] += red[tid + s]; __syncthreads(); }
    const float scale_q = rsqrtf(red[0] / (float)QLORA_ + EPS_);
    __syncthreads();
    for (int i = tid; i < QLORA_; i += 256)
        q_norm[b * QLORA_ + i] = qa_raw[b * QLORA_ + i] * scale_q * gamma_q[i];

    // rms(c_kv)
    float ss2 = 0.f;
    for (int i = tid; i < KVLORA_; i += 256) {
        float v = ckv_raw[b * CW_ + i];
        ss2 += v * v;
    }
    red[tid] = ss2;
    __syncthreads();
    for (int s = 128; s; s >>= 1) { if (tid < s) red[tid] += red[tid + s]; __syncthreads(); }
    const float scale_kv = rsqrtf(red[0] / (float)KVLORA_ + EPS_);

    float* new_row = cache_out + ((size_t)b * TTOT_ + PAST_) * CW_;
    for (int i = tid; i < KVLORA_; i += 256)
        new_row[i] = ckv_raw[b * CW_ + i] * scale_kv * gamma_kv[i];

    // RoPE(k_pe)
    const long long pos = pos_ids[b];
    if (tid < DROPE_) {
        float c = cosT[pos * DROPE_ + tid];
        float s = sinT[pos * DROPE_ + tid];
        float v = ckv_raw[b * CW_ + KVLORA_ + tid];
        float rot = (tid < DROPE_ / 2) ? -ckv_raw[b * CW_ + KVLORA_ + tid + DROPE_ / 2]
                                       :  ckv_raw[b * CW_ + KVLORA_ + tid - DROPE_ / 2];
        new_row[KVLORA_ + tid] = v * c + rot * s;
    }
}

// ---------------------------------------------------------------------------
// Stage 3: q_raw = q_norm @ w_qb.T    (one wave per output row element)
// ---------------------------------------------------------------------------
__global__ void k_qb(const float* __restrict__ q_norm,
                     const float* __restrict__ w_qb,
                     float* __restrict__ q_raw) {
    const int wid  = (blockIdx.x * blockDim.x + threadIdx.x) >> 5;
    const int lane = threadIdx.x & 31;
    const int b = wid / (H_ * DQK_);
    const int r = wid % (H_ * DQK_);
    if (b >= B_) return;
    const float* qn  = q_norm + b * QLORA_;
    const float* row = w_qb + (size_t)r * QLORA_;
    float acc = 0.f;
    for (int i = lane * 4; i < QLORA_; i += 128) {
        __builtin_prefetch(row + i + 1024, 0, 0);
        float4 a = *(const float4*)(qn + i);
        float4 w = *(const float4*)(row + i);
        acc += a.x * w.x + a.y * w.y + a.z * w.z + a.w * w.w;
    }
    acc = wave_sum(acc);
    if (lane == 0) q_raw[(size_t)b * H_ * DQK_ + r] = acc;
}

// ---------------------------------------------------------------------------
// Stage 4: MLA absorption — q_full[b,h,0:512] = q_nope @ w_kvb_k[h],
//          q_full[b,h,512:576] = RoPE(q_pe); both pre-scaled by 1/sqrt(192)
// ---------------------------------------------------------------------------
__global__ void k_absorb(const float* __restrict__ q_raw,
                         const float* __restrict__ w_kvb,
                         const long long* __restrict__ pos_ids,
                         const float* __restrict__ cosT,
                         const float* __restrict__ sinT,
                         float* __restrict__ q_full) {
    const int h = blockIdx.x;
    const int tid = threadIdx.x;            // 512 threads
    __shared__ float qn[B_][DNOPE_];
    {
        int b = tid >> 7, d = tid & 127;
        qn[b][d] = q_raw[((size_t)b * H_ + h) * DQK_ + d];
    }
    const float inv_scale = rsqrtf((float)DQK_);
    if (tid < B_ * DROPE_) {
        int b = tid / DROPE_, i = tid % DROPE_;
        const long long pos = pos_ids[b];
        float c = cosT[pos * DROPE_ + i];
        float s = sinT[pos * DROPE_ + i];
        const float* qp = q_raw + ((size_t)b * H_ + h) * DQK_ + DNOPE_;
        float v = qp[i];
        float rot = (i < DROPE_ / 2) ? -qp[i + DROPE_ / 2] : qp[i - DROPE_ / 2];
        q_full[((size_t)b * H_ + h) * CW_ + KVLORA_ + i] = (v * c + rot * s) * inv_scale;
    }
    __syncthreads();

    const int c = tid;                      // 0..511 latent column
    float a0 = 0.f, a1 = 0.f, a2 = 0.f, a3 = 0.f;
    const float* wk = w_kvb + (size_t)h * 256 * KVLORA_;   // k_nope rows of head h
#pragma unroll 4
    for (int d = 0; d < DNOPE_; ++d) {
        float w = wk[(size_t)d * KVLORA_ + c];
        a0 += qn[0][d] * w; a1 += qn[1][d] * w;
        a2 += qn[2][d] * w; a3 += qn[3][d] * w;
    }
    q_full[((size_t)0 * H_ + h) * CW_ + c] = a0 * inv_scale;
    q_full[((size_t)1 * H_ + h) * CW_ + c] = a1 * inv_scale;
    q_full[((size_t)2 * H_ + h) * CW_ + c] = a2 * inv_scale;
    q_full[((size_t)3 * H_ + h) * CW_ + c] = a3 * inv_scale;
}

// ---------------------------------------------------------------------------
// Stage 5: WMMA attention in latent space.
//   block = (batch b, 16-head tile). 4 waves.
//   scores[16h x 1025t] = qA[16 x 576] x cache^T      (V_WMMA_F32_16X16X4_F32)
//   softmax rows in LDS
//   ctxl[16h x 512c]   = P[16 x 1025] x cache          (V_WMMA_F32_16X16X4_F32)
//   ctx[b,h,dv]        = ctxl @ w_kvb_v[h]^T
//   Hot loops carry no boundary selects; guarded tail tiles handled separately.
// ---------------------------------------------------------------------------
__global__ void __launch_bounds__(128)
k_attn(const float* __restrict__ q_full,
       const float* __restrict__ cache,
       const float* __restrict__ mask,
       const float* __restrict__ w_kvb,
       float* __restrict__ ctx) {
    const int b    = blockIdx.y;
    const int h0   = blockIdx.x * 16;
    const int tid  = threadIdx.x;
    const int wave = tid >> 5;
    const int lane = tid & 31;
    const int m    = lane & 15;            // tile row / col index for this lane
    const int kp   = (lane >> 4) * 2;      // k-pair base for this lane

    __shared__ float qA[16][CW_ + 4];      // 16 x 580 (row stride 2320B, 16B aligned)
    __shared__ float sc[16][1056];         // scores / probabilities, K padded
    __shared__ float ctxl[16][KVLORA_ + 4];

#ifdef ASYNC_LDS_OK
    // stage q tile via async global->LDS (ASYNCcnt), 16B chunks
    {
        const float* src = q_full + ((size_t)b * H_ + h0) * CW_;
        for (int i = tid; i < 16 * (CW_ / 4); i += 128) {
            const int r  = i / (CW_ / 4);
            const int c4 = (i % (CW_ / 4)) * 4;
            __builtin_amdgcn_global_load_async_to_lds_b128(
                (__attribute__((address_space(1))) b128_t*)(src + (size_t)r * CW_ + c4),
                (__attribute__((address_space(3))) b128_t*)&qA[r][c4],
                0, 0);
        }
        __builtin_amdgcn_s_wait_asynccnt(0);
    }
#else
    for (int i = tid; i < 16 * CW_; i += 128) {
        int r = i / CW_, c = i % CW_;
        qA[r][c] = q_full[((size_t)b * H_ + h0 + r) * CW_ + c];
    }
#endif
    for (int i = tid; i < 16 * 3; i += 128)        // zero K-pad cols 1025..1027
        sc[i / 3][TTOT_ + (i % 3)] = 0.f;
    __syncthreads();

    const float* cb = cache + (size_t)b * TTOT_ * CW_;

    // ---- scores GEMM main: tiles 0..63 (all t valid), no guards ----
    for (int nt = wave; nt < 64; nt += 4) {
        const int t = nt * 16 + m;
        const float* bp = cb + (size_t)t * CW_ + kp;
        v8f acc = {0.f, 0.f, 0.f, 0.f, 0.f, 0.f, 0.f, 0.f};
        for (int k = 0; k < CW_; k += 4) {
            v2f a;  a.x  = qA[m][k + kp];  a.y  = qA[m][k + kp + 1];
            v2f bb; bb.x = bp[k];          bb.y = bp[k + 1];
            acc = __builtin_amdgcn_wmma_f32_16x16x4_f32(
                false, a, false, bb, (short)0, acc, false, false);
        }
#pragma unroll
        for (int i = 0; i < 8; ++i)
            sc[i + (lane >> 4) * 8][t] = acc[i];
    }
    // ---- scores GEMM tail: tile 64 (tbase=1024, only m==0 valid) ----
    if (wave == 0) {
        const int t = 1024 + m;
        const float tv = (t < TTOT_) ? 1.f : 0.f;
        const int tc = (t < TTOT_) ? t : (TTOT_ - 1);
        const float* bp = cb + (size_t)tc * CW_ + kp;
        v8f acc = {0.f, 0.f, 0.f, 0.f, 0.f, 0.f, 0.f, 0.f};
        for (int k = 0; k < CW_; k += 4) {
            v2f a;  a.x  = qA[m][k + kp];  a.y  = qA[m][k + kp + 1];
            v2f bb; bb.x = bp[k] * tv;     bb.y = bp[k + 1] * tv;
            acc = __builtin_amdgcn_wmma_f32_16x16x4_f32(
                false, a, false, bb, (short)0, acc, false, false);
        }
#pragma unroll
        for (int i = 0; i < 8; ++i) {
            int row = i + (lane >> 4) * 8;
            if (t < TTOT_) sc[row][t] = acc[i];
        }
    }
    __syncthreads();

    // ---- softmax (each wave owns 4 head rows) ----
    const float* mk = mask + (size_t)b * TTOT_;
    for (int r = wave * 4; r < wave * 4 + 4; ++r) {
        float mx = -3.4e38f;
        for (int t = lane; t < TTOT_; t += 32) mx = fmaxf(mx, sc[r][t] + mk[t]);
        mx = wave_max(mx);
        float sum = 0.f;
        for (int t = lane; t < TTOT_; t += 32) {
            float e = __expf(sc[r][t] + mk[t] - mx);
            sc[r][t] = e;
            sum += e;
        }
        sum = wave_sum(sum);
        const float inv = 1.f / sum;
        for (int t = lane; t < TTOT_; t += 32) sc[r][t] *= inv;
    }
    __syncthreads();

    // ---- ctx GEMM: M=16, N=512 (tiles of 16), K=1025 (main 1024 + tail) ----
    for (int nt = wave; nt < KVLORA_ / 16; nt += 4) {
        const int n = nt * 16 + m;
        const float* bpn = cb + n;
        v8f acc = {0.f, 0.f, 0.f, 0.f, 0.f, 0.f, 0.f, 0.f};
        for (int k = 0; k < 1024; k += 4) {
            const int t0 = k + kp;
            v2f a;  a.x  = sc[m][t0];
                    a.y  = sc[m][t0 + 1];
            v2f bb; bb.x = bpn[(size_t)t0 * CW_];
                    bb.y = bpn[(size_t)(t0 + 1) * CW_];
            acc = __builtin_amdgcn_wmma_f32_16x16x4_f32(
                false, a, false, bb, (short)0, acc, false, false);
        }
        {   // K tail: k = 1024 (t in 1024..1027; sc pad cols are zero)
            const int t0 = 1024 + kp, t1 = t0 + 1;
            const int t0c = (t0 < TTOT_) ? t0 : (TTOT_ - 1);
            const int t1c = (t1 < TTOT_) ? t1 : (TTOT_ - 1);
            v2f a;  a.x = sc[m][t0];  a.y = sc[m][t1];
            v2f bb;
            bb.x = bpn[(size_t)t0c * CW_] * ((t0 < TTOT_) ? 1.f : 0.f);
            bb.y = bpn[(size_t)t1c * CW_] * ((t1 < TTOT_) ? 1.f : 0.f);
            acc = __builtin_amdgcn_wmma_f32_16x16x4_f32(
                false, a, false, bb, (short)0, acc, false, false);
        }
#pragma unroll
        for (int i = 0; i < 8; ++i)
            ctxl[i + (lane >> 4) * 8][n] = acc[i];
    }
    __syncthreads();

    // ---- value projection: ctx[b,h0+r,dv] = ctxl[r,:] . w_kvb_v[h0+r][dv,:] ----
    for (int i = tid; i < 16 * DV_; i += 128) {
        const int r = i / DV_, dv = i % DV_;
        const float* wv = w_kvb + ((size_t)(h0 + r) * 256 + DNOPE_ + dv) * KVLORA_;
        float acc = 0.f;
        for (int c = 0; c < KVLORA_; c += 4) {
            acc += ctxl[r][c]     * wv[c]
                 + ctxl[r][c + 1] * wv[c + 1]
                 + ctxl[r][c + 2] * wv[c + 2]
                 + ctxl[r][c + 3] * wv[c + 3];
        }
        ctx[((size_t)b * H_ + h0 + r) * DV_ + dv] = acc;
    }
}

// ---------------------------------------------------------------------------
// Stage 6: attn_out = ctx @ w_o.T   (one wave per output col, 4 batches at once)
// ---------------------------------------------------------------------------
__global__ void k_out(const float* __restrict__ ctx,
                      const float* __restrict__ w_o,
                      float* __restrict__ out) {
    const int wid  = (blockIdx.x * blockDim.x + threadIdx.x) >> 5;
    const int lane = threadIdx.x & 31;
    if (wid >= HID_) return;
    const float* row = w_o + (size_t)wid * (H_ * DV_);
    float a0 = 0.f, a1 = 0.f, a2 = 0.f, a3 = 0.f;
    for (int j = lane * 4; j < H_ * DV_; j += 128) {
        __builtin_prefetch(row + j + 2048, 0, 0);
        float4 w  = *(const float4*)(row + j);
        float4 c0 = *(const float4*)(ctx + 0 * H_ * DV_ + j);
        float4 c1 = *(const float4*)(ctx + 1 * H_ * DV_ + j);
        float4 c2 = *(const float4*)(ctx + 2 * H_ * DV_ + j);
        float4 c3 = *(const float4*)(ctx + 3 * H_ * DV_ + j);
        a0 += w.x * c0.x + w.y * c0.y + w.z * c0.z + w.w * c0.w;
        a1 += w.x * c1.x + w.y * c1.y + w.z * c1.z + w.w * c1.w;
        a2 += w.x * c2.x + w.y * c2.y + w.z * c2.z + w.w * c2.w;
        a3 += w.x * c3.x + w.y * c3.y + w.z * c3.z + w.w * c3.w;
    }
    a0 = wave_sum(a0); a1 = wave_sum(a1); a2 = wave_sum(a2); a3 = wave_sum(a3);
    if (lane == 0) {
        out[0 * HID_ + wid] = a0;
        out[1 * HID_ + wid] = a1;
        out[2 * HID_ + wid] = a2;
        out[3 * HID_ + wid] = a3;
    }
}

// ---------------------------------------------------------------------------
extern "C" void kernel_launch(void* const* d_in, const int* in_sizes, int n_in,
                              void* d_out, int out_size, void* d_ws, size_t ws_size,
                              hipStream_t stream) {
    (void)in_sizes; (void)n_in; (void)out_size; (void)ws_size;

    const float*     x        = (const float*)d_in[0];
    const float*     mask     = (const float*)d_in[1];
    const long long* pos_ids  = (const long long*)d_in[2];
    const float*     past     = (const float*)d_in[3];
    const float*     cosT     = (const float*)d_in[4];
    const float*     sinT     = (const float*)d_in[5];
    const float*     w_qa     = (const float*)d_in[6];
    const float*     gamma_q  = (const float*)d_in[7];
    const float*     w_qb     = (const float*)d_in[8];
    const float*     w_kva    = (const float*)d_in[9];
    const float*     gamma_kv = (const float*)d_in[10];
    const float*     w_kvb    = (const float*)d_in[11];
    const float*     w_o      = (const float*)d_in[12];

    float* attn_out  = (float*)d_out;                         // B*HID
    float* cache_out = attn_out + (size_t)B_ * HID_;          // B*1025*576

    float* ws      = (float*)d_ws;
    float* qa_raw  = ws;  ws += (size_t)B_ * QLORA_;
    float* ckv_raw = ws;  ws += (size_t)B_ * CW_;
    float* q_norm  = ws;  ws += (size_t)B_ * QLORA_;
    float* q_raw   = ws;  ws += (size_t)B_ * H_ * DQK_;
    float* q_full  = ws;  ws += (size_t)B_ * H_ * CW_;
    float* ctx     = ws;  ws += (size_t)B_ * H_ * DV_;

    // past cache -> output cache rows [0,1024)
    for (int b = 0; b < B_; ++b) {
        (void)hipMemcpyAsync(cache_out + (size_t)b * TTOT_ * CW_,
                             past + (size_t)b * PAST_ * CW_,
                             (size_t)PAST_ * CW_ * sizeof(float),
                             hipMemcpyDeviceToDevice, stream);
    }

    k_gemv_qa_kva<<<(B_ * (QLORA_ + CW_)) / 8, 256, 0, stream>>>(
        x, w_qa, w_kva, qa_raw, ckv_raw);
    k_norm_rope_cache<<<B_, 256, 0, stream>>>(
        qa_raw, ckv_raw, gamma_q, gamma_kv, pos_ids, cosT, sinT, q_norm, cache_out);
    k_qb<<<(B_ * H_ * DQK_) / 8, 256, 0, stream>>>(q_norm, w_qb, q_raw);
    k_absorb<<<H_, 512, 0, stream>>>(q_raw, w_kvb, pos_ids, cosT, sinT, q_full);
    dim3 ga(H_ / 16, B_);
    k_attn<<<ga, 128, 0, stream>>>(q_full, cache_out, mask, w_kvb, ctx);
    k_out<<<HID_ / 8, 256, 0, stream>>>(ctx, w_o, attn_out);
}